// HDLUT_66108136620278
// MI455X (gfx1250) — compile-verified
//
#include <hip/hip_runtime.h>

// HDLUT 2x super-resolution, single fused pass.
// For LR pixel (y,x): 8 table gathers (h: R,D,L,U ; d: DR,DL,UL,UR), each a
// float4 row of W[c*256+nb]; round-r rotation becomes a fixed permutation of
// the 2x2 output block. Output = sum/2 written directly at (2y+s, 2x+t).

#define LRN   512           // LR image side
#define HRN   1024          // HR image side
#define TX    32            // tile width  (LR px)
#define TY    32            // tile height (LR px)
#define TW    (TX + 2)      // tile + halo
#define TH    (TY + 2)
#define NTHR  256

typedef float v2f __attribute__((ext_vector_type(2)));
typedef float v4f __attribute__((ext_vector_type(4)));

__global__ __launch_bounds__(NTHR)
void hdlut_sr2x_kernel(const int* __restrict__ img,
                       const v4f* __restrict__ hw,   // 65536 x float4
                       const v4f* __restrict__ dw,   // 65536 x float4
                       float* __restrict__ out)
{
    __shared__ int tile[TH][TW];

    const int plane = blockIdx.z;                 // B*C planes
    const int ty0   = blockIdx.y * TY;
    const int tx0   = blockIdx.x * TX;
    const int tid   = threadIdx.x;

    const int* img_p = img + (size_t)plane * (LRN * LRN);

    // ---- Fill LDS tile (clamped 1-px halo) with CDNA5 async global->LDS ----
    for (int e = tid; e < TH * TW; e += NTHR) {
        int r = e / TW;
        int c = e - r * TW;
        int gy = ty0 - 1 + r; gy = gy < 0 ? 0 : (gy > LRN - 1 ? LRN - 1 : gy);
        int gx = tx0 - 1 + c; gx = gx < 0 ? 0 : (gx > LRN - 1 ? LRN - 1 : gx);
        unsigned lds_off = (unsigned)(uintptr_t)&tile[r][c];   // low 32b = LDS addr
        unsigned voff    = (unsigned)((gy * LRN + gx) * 4);
        asm volatile("global_load_async_to_lds_b32 %0, %1, %2"
                     :: "v"(lds_off), "v"(voff), "s"(img_p)
                     : "memory");
    }
    asm volatile("s_wait_asynccnt 0x0" ::: "memory");
    __syncthreads();

    // ---- Compute: 4 LR pixels per thread (32 wide x 8 rows x 4 passes) ----
    const int lx  = tid & 31;
    const int ly0 = tid >> 5;
    const size_t oplane = (size_t)plane * ((size_t)HRN * HRN);

    #pragma unroll
    for (int k = 0; k < 4; ++k) {
        const int ly = ly0 + k * 8;
        const int cy = ly + 1, cx = lx + 1;

        const int c  = tile[cy][cx];
        const int up = tile[cy - 1][cx],     dn = tile[cy + 1][cx];
        const int lf = tile[cy][cx - 1],     rt = tile[cy][cx + 1];
        const int ul = tile[cy - 1][cx - 1], ur = tile[cy - 1][cx + 1];
        const int dl = tile[cy + 1][cx - 1], dr = tile[cy + 1][cx + 1];
        const int base = c << 8;

        const v4f hR  = hw[base + rt];   // h, r=0
        const v4f hD  = hw[base + dn];   // h, r=1
        const v4f hL  = hw[base + lf];   // h, r=2
        const v4f hU  = hw[base + up];   // h, r=3
        const v4f dDR = dw[base + dr];   // d, r=0
        const v4f dDL = dw[base + dl];   // d, r=1
        const v4f dUL = dw[base + ul];   // d, r=2
        const v4f dUR = dw[base + ur];   // d, r=3

        // Per-round element permutation (derived from rot90 algebra):
        // r0: e=2s+t  r1: e=2(1-t)+s  r2: e=2(1-s)+(1-t)  r3: e=2t+(1-s)
        const float o00 = (hR.x + dDR.x) + (hD.z + dDL.z) + (hL.w + dUL.w) + (hU.y + dUR.y);
        const float o01 = (hR.y + dDR.y) + (hD.x + dDL.x) + (hL.z + dUL.z) + (hU.w + dUR.w);
        const float o10 = (hR.z + dDR.z) + (hD.w + dDL.w) + (hL.y + dUL.y) + (hU.x + dUR.x);
        const float o11 = (hR.w + dDR.w) + (hD.y + dDL.y) + (hL.x + dUL.x) + (hU.z + dUR.z);

        const int Y = ty0 + ly, X = tx0 + lx;
        v2f* o0 = (v2f*)(out + oplane + (size_t)(2 * Y)     * HRN) + X;
        v2f* o1 = (v2f*)(out + oplane + (size_t)(2 * Y + 1) * HRN) + X;
        // Non-temporal: 201 MB write-once stream must not evict hot LUTs in L2.
        v2f r0v = { o00 * 0.5f, o01 * 0.5f };
        v2f r1v = { o10 * 0.5f, o11 * 0.5f };
        __builtin_nontemporal_store(r0v, o0);
        __builtin_nontemporal_store(r1v, o1);
    }
}

extern "C" void kernel_launch(void* const* d_in, const int* in_sizes, int n_in,
                              void* d_out, int out_size, void* d_ws, size_t ws_size,
                              hipStream_t stream) {
    (void)n_in; (void)out_size; (void)d_ws; (void)ws_size;
    const int* img = (const int*)d_in[0];
    const v4f* hw  = (const v4f*)d_in[1];
    const v4f* dw  = (const v4f*)d_in[2];
    float*     out = (float*)d_out;

    const int planes = in_sizes[0] / (LRN * LRN);   // B*C = 48
    dim3 grid(LRN / TX, LRN / TY, planes);
    hdlut_sr2x_kernel<<<grid, NTHR, 0, stream>>>(img, hw, dw, out);
}